// NarrativeGraph_9414568313175
// MI455X (gfx1250) — compile-verified
//
#include <hip/hip_runtime.h>
#include <math.h>

// Problem constants (from reference)
#define H    128
#define R_   8
#define B_   2
#define T_   64
#define TWO_H 256
#define THREE_H 384
#define LN_EPS 1e-5f

typedef float v2f __attribute__((ext_vector_type(2)));
typedef float v8f __attribute__((ext_vector_type(8)));

// Workspace layout (floats):
//   [0 .. 4096)            : S sums, S[r][b][k] at r*512 + b*256 + k  (k in [0,2H))
//   [4096 .. 4104)         : per-relation edge counts (as float)
//   [4104 .. 4104+128*384) : gx[m][j], m = b*64 + t
#define WS_S      0
#define WS_CNT    4096
#define WS_GX     4104
#define WS_ZERO_N 4104

// ---------------------------------------------------------------- kernel 1
__global__ void ng_zero_ws(float* ws) {
    int i = blockIdx.x * blockDim.x + threadIdx.x;
    if (i < WS_ZERO_N) ws[i] = 0.0f;
}

// ---------------------------------------------------------------- kernel 2
// Per-relation gather-accumulate. One edge per wave per iteration.
// Inner loop is pure VALU: per-lane register accumulators acc[r][16]
// (8 src + 8 dst elements per lane), relation selected by an unrolled
// wave-uniform compare. No LDS traffic per edge; single LDS+global
// atomic flush per block at the end.
__global__ void ng_edge_accum(const float* __restrict__ nodes,
                              const int* __restrict__ src,
                              const int* __restrict__ rel,
                              const int* __restrict__ dst,
                              int E, float* __restrict__ gS,
                              float* __restrict__ gCnt) {
    __shared__ float acc_s[R_ * 512];
    __shared__ float cnt_s[R_];
    const int tid  = threadIdx.x;
    const int wave = tid >> 5;
    const int lane = tid & 31;

    for (int i = tid; i < R_ * 512; i += blockDim.x) acc_s[i] = 0.0f;
    if (tid < R_) cnt_s[tid] = 0.0f;
    __syncthreads();

    // register accumulators: acc[r][0..7] = src slice, acc[r][8..15] = dst slice
    float acc[R_][16];
    float cntw[R_];
#pragma unroll
    for (int r = 0; r < R_; ++r) {
        cntw[r] = 0.0f;
#pragma unroll
        for (int c = 0; c < 16; ++c) acc[r][c] = 0.0f;
    }

    const int wavesPerBlk = blockDim.x >> 5;
    const int base = lane * 8;        // 32 lanes * 8 floats = 256 = B*H
    const int off  = base & 128;      // [B,H] flat -> [B,2H] slot remap

    for (int e = blockIdx.x * wavesPerBlk + wave; e < E;
         e += gridDim.x * wavesPerBlk) {
        const int r = rel[e];
        const float* ps = nodes + (size_t)src[e] * 256 + base;
        const float* pd = nodes + (size_t)dst[e] * 256 + base;
        float4 s0 = *(const float4*)(ps);
        float4 s1 = *(const float4*)(ps + 4);
        float4 d0 = *(const float4*)(pd);
        float4 d1 = *(const float4*)(pd + 4);
        float v[16] = { s0.x, s0.y, s0.z, s0.w, s1.x, s1.y, s1.z, s1.w,
                        d0.x, d0.y, d0.z, d0.w, d1.x, d1.y, d1.z, d1.w };
#pragma unroll
        for (int rr = 0; rr < R_; ++rr) {
            if (r == rr) {              // wave-uniform predicate
#pragma unroll
                for (int c = 0; c < 16; ++c) acc[rr][c] += v[c];
                cntw[rr] += 1.0f;
            }
        }
    }

    // flush registers -> block LDS accumulator (cross-wave collisions -> ds atomics)
#pragma unroll
    for (int rr = 0; rr < R_; ++rr) {
        float* a = acc_s + rr * 512 + base + off;
#pragma unroll
        for (int c = 0; c < 8; ++c) atomicAdd(&a[c], acc[rr][c]);
#pragma unroll
        for (int c = 0; c < 8; ++c) atomicAdd(&a[128 + c], acc[rr][8 + c]);
    }
    if (lane == 0) {
#pragma unroll
        for (int rr = 0; rr < R_; ++rr) atomicAdd(&cnt_s[rr], cntw[rr]);
    }
    __syncthreads();

    // flush block LDS -> global
    for (int i = tid; i < R_ * 512; i += blockDim.x) atomicAdd(&gS[i], acc_s[i]);
    if (tid < R_) atomicAdd(&gCnt[tid], cnt_s[tid]);
}

// ---------------------------------------------------------------- kernel 3
// gx[m][j] = sum_k X[m][k] * Wih[j][k], m in [0,128), j in [0,384), k in [0,128)
// fp32 WMMA 16x16x4, one wave per 16x16 output tile. 8 x 24 = 192 waves.
__global__ void ng_gx_wmma(const float* __restrict__ X,     // [128,128]
                           const float* __restrict__ Wih,   // [384,128]
                           float* __restrict__ gx) {        // [128,384]
    const int wave = (blockIdx.x * blockDim.x + threadIdx.x) >> 5;
    const int lane = threadIdx.x & 31;
    const int mt = wave / 24;          // 0..7
    const int nt = wave % 24;          // 0..23
    const int mrow = mt * 16 + (lane & 15);   // A: M = lane%16
    const int ncol = nt * 16 + (lane & 15);   // B: N = lane%16
    const int koff = (lane >> 4) << 1;        // VGPR K offset: 0 or 2

    v8f c = {0.f, 0.f, 0.f, 0.f, 0.f, 0.f, 0.f, 0.f};
    const float* ap = X   + mrow * H;
    const float* bp = Wih + ncol * H;   // B[k][n] = Wih[n][k]
#pragma unroll 4
    for (int kk = 0; kk < H; kk += 4) {
        float2 at = *(const float2*)(ap + kk + koff);
        float2 bt = *(const float2*)(bp + kk + koff);
        v2f a; a.x = at.x; a.y = at.y;
        v2f b; b.x = bt.x; b.y = bt.y;
        c = __builtin_amdgcn_wmma_f32_16x16x4_f32(
                false, a, false, b, (short)0, c, false, false);
    }
    // D layout: VGPR i holds rows mt*16+i (lanes 0-15) and mt*16+8+i (lanes 16-31)
    const int mbase = mt * 16 + ((lane >> 4) << 3);
#pragma unroll
    for (int i = 0; i < 8; ++i)
        gx[(mbase + i) * THREE_H + ncol] = c[i];
}

// ---------------------------------------------------------------- kernel 4
// Single block, 768 threads: relation matmuls + mean, GRU scan, MLP+LN.
__global__ void __launch_bounds__(768)
ng_fused_tail(const float* __restrict__ gS,      // [8][2][256]
              const float* __restrict__ gCnt,    // [8]
              const float* __restrict__ gx,      // [128][384]
              const float* __restrict__ rel_W,   // [8][256][128]
              const float* __restrict__ rel_b,   // [8][128]
              const float* __restrict__ Whh,     // [384][128]
              const float* __restrict__ bih,     // [384]
              const float* __restrict__ bhh,     // [384]
              const float* __restrict__ W1,      // [128][256]
              const float* __restrict__ b1,      // [256]
              const float* __restrict__ ln_g,    // [256]
              const float* __restrict__ ln_b,    // [256]
              const float* __restrict__ W2,      // [256][128]
              const float* __restrict__ b2,      // [128]
              float E_f,
              float* __restrict__ out) {         // [2][128]
    __shared__ float s_gs[B_ * H];        // graph_state, then combined state
    __shared__ float s_h[B_ * H];         // GRU hidden
    __shared__ float s_gh[B_ * THREE_H];  // GRU hidden-gate preacts
    __shared__ float s_h1[B_ * TWO_H];    // MLP hidden
    __shared__ float s_mu[B_], s_rs[B_];

    const int tid = threadIdx.x;

    // ---- phase A: graph_state = (sum_r S_r @ W_r + cnt_r * b_r) / E
    if (tid < B_ * H) {
        const int b = tid >> 7, i = tid & 127;
        float acc = 0.0f;
        for (int r = 0; r < R_; ++r) {
            acc += gCnt[r] * rel_b[r * H + i];
            const float* s = gS + r * 512 + b * TWO_H;
            const float* w = rel_W + (size_t)r * TWO_H * H + i;
            for (int k = 0; k < TWO_H; ++k)
                acc = fmaf(s[k], w[(size_t)k * H], acc);
        }
        s_gs[tid] = acc / E_f;
        s_h[tid] = 0.0f;
    }
    __syncthreads();

    // ---- phase B: GRU scan over T=64 steps
    for (int t = 0; t < T_; ++t) {
        {   // gh[b][j] = h[b] . Whh[j] + bhh[j]   (768 dots, one per thread)
            const int b = tid / THREE_H, j = tid - b * THREE_H;
            const float* hp = s_h + b * H;
            const float* wp = Whh + j * H;
            float acc = bhh[j];
            for (int k = 0; k < H; ++k) acc = fmaf(hp[k], wp[k], acc);
            s_gh[tid] = acc;
        }
        __syncthreads();
        if (tid < B_ * H) {
            const int b = tid >> 7, i = tid & 127;
            const int m = b * T_ + t;
            const float* g = gx + m * THREE_H;
            float xr = g[i]           + bih[i];
            float xz = g[H + i]       + bih[H + i];
            float xn = g[2 * H + i]   + bih[2 * H + i];
            float hr = s_gh[b * THREE_H + i];
            float hz = s_gh[b * THREE_H + H + i];
            float hn = s_gh[b * THREE_H + 2 * H + i];
            float r = 1.0f / (1.0f + expf(-(xr + hr)));
            float z = 1.0f / (1.0f + expf(-(xz + hz)));
            float n = tanhf(xn + r * hn);
            s_h[tid] = (1.0f - z) * n + z * s_h[tid];
        }
        __syncthreads();
    }

    // ---- phase C: combine
    if (tid < B_ * H) s_gs[tid] += s_h[tid];
    __syncthreads();

    // ---- phase D: h1 = gs @ W1 + b1   (512 outputs)
    if (tid < B_ * TWO_H) {
        const int b = tid >> 8, j = tid & 255;
        const float* s = s_gs + b * H;
        float acc = b1[j];
        for (int k = 0; k < H; ++k) acc = fmaf(s[k], W1[k * TWO_H + j], acc);
        s_h1[tid] = acc;
    }
    __syncthreads();

    // ---- phase E: LayerNorm + ReLU
    if (tid < B_) {
        const float* x = s_h1 + tid * TWO_H;
        float mu = 0.0f;
        for (int k = 0; k < TWO_H; ++k) mu += x[k];
        mu *= (1.0f / TWO_H);
        float var = 0.0f;
        for (int k = 0; k < TWO_H; ++k) { float d = x[k] - mu; var = fmaf(d, d, var); }
        var *= (1.0f / TWO_H);
        s_mu[tid] = mu;
        s_rs[tid] = rsqrtf(var + LN_EPS);
    }
    __syncthreads();
    if (tid < B_ * TWO_H) {
        const int b = tid >> 8, j = tid & 255;
        float v = (s_h1[tid] - s_mu[b]) * s_rs[b] * ln_g[j] + ln_b[j];
        s_h1[tid] = fmaxf(v, 0.0f);
    }
    __syncthreads();

    // ---- phase F: out = h1 @ W2 + b2
    if (tid < B_ * H) {
        const int b = tid >> 7, i = tid & 127;
        const float* hp = s_h1 + b * TWO_H;
        float acc = b2[i];
        for (int k = 0; k < TWO_H; ++k) acc = fmaf(hp[k], W2[k * H + i], acc);
        out[b * H + i] = acc;
    }
}

// ---------------------------------------------------------------- launcher
extern "C" void kernel_launch(void* const* d_in, const int* in_sizes, int n_in,
                              void* d_out, int out_size, void* d_ws, size_t ws_size,
                              hipStream_t stream) {
    const float* nodes    = (const float*)d_in[0];
    const float* temporal = (const float*)d_in[1];
    const float* rel_W    = (const float*)d_in[2];
    const float* rel_b    = (const float*)d_in[3];
    const float* gru_Wih  = (const float*)d_in[4];
    const float* gru_Whh  = (const float*)d_in[5];
    const float* gru_bih  = (const float*)d_in[6];
    const float* gru_bhh  = (const float*)d_in[7];
    const float* mlp_W1   = (const float*)d_in[8];
    const float* mlp_b1   = (const float*)d_in[9];
    const float* ln_g     = (const float*)d_in[10];
    const float* ln_b     = (const float*)d_in[11];
    const float* mlp_W2   = (const float*)d_in[12];
    const float* mlp_b2   = (const float*)d_in[13];
    const int*   src      = (const int*)d_in[14];
    const int*   rel      = (const int*)d_in[15];
    const int*   dst      = (const int*)d_in[16];
    const int E = in_sizes[14];

    float* ws   = (float*)d_ws;
    float* gS   = ws + WS_S;
    float* gCnt = ws + WS_CNT;
    float* gx   = ws + WS_GX;
    float* out  = (float*)d_out;

    ng_zero_ws<<<(WS_ZERO_N + 255) / 256, 256, 0, stream>>>(ws);

    ng_edge_accum<<<160, 256, 0, stream>>>(nodes, src, rel, dst, E, gS, gCnt);

    // 192 waves = 24 blocks x 8 waves; exact fit, no divergent guard (EXEC all-1)
    ng_gx_wmma<<<24, 256, 0, stream>>>(temporal, gru_Wih, gx);

    ng_fused_tail<<<1, 768, 0, stream>>>(gS, gCnt, gx, rel_W, rel_b,
                                         gru_Whh, gru_bih, gru_bhh,
                                         mlp_W1, mlp_b1, ln_g, ln_b,
                                         mlp_W2, mlp_b2, (float)E, out);
}